// SOM_39221641347646
// MI455X (gfx1250) — compile-verified
//
#include <hip/hip_runtime.h>
#include <hip/hip_bf16.h>
#include <stdint.h>

typedef __attribute__((ext_vector_type(2))) float v2f;
typedef __attribute__((ext_vector_type(8))) float v8f;

#define DIMS        128
#define TILE        16
#define ROWPAD      132          // 128 + 4 floats: kills LDS bank conflicts on strided reads
#define K1_BLOCKS   2048
#define K1_THREADS  64           // 2 waves (wave32); double-buffered slabs fit in LDS
#define K1_WAVES    (K1_THREADS / 32)
#define K2_THREADS  256
#define TOPK        16
#define FMAXV       3.402823466e+38f

// ---- CDNA5 async global->LDS copy (ASYNCcnt pipe, no VGPR staging) ----------
__device__ __forceinline__ void async_copy_b128(uint32_t lds_byte_off, const float* gaddr) {
    asm volatile("global_load_async_to_lds_b128 %0, %1, off"
                 :: "v"(lds_byte_off), "v"(gaddr) : "memory");
}
__device__ __forceinline__ void wait_async_le16() {
    asm volatile("s_wait_asynccnt 0x10" ::: "memory");
}
__device__ __forceinline__ void wait_async_0() {
    asm volatile("s_wait_asynccnt 0x0" ::: "memory");
}
__device__ __forceinline__ void wait_ds_0() {
    asm volatile("s_wait_dscnt 0x0" ::: "memory");
}

// Insert (d2, idx) into an ascending sorted register list. Static indexing only
// (fully unrolled) so it never spills to scratch. Early-out on the list max.
__device__ __forceinline__ void topk_insert(float (&bd)[TOPK], int (&bi)[TOPK],
                                            float d2, int idx) {
    if (d2 < bd[TOPK - 1]) {
        #pragma unroll
        for (int j = TOPK - 1; j >= 1; --j) {
            if (bd[j] > d2) {
                bool shift = (bd[j - 1] > d2);
                bd[j] = shift ? bd[j - 1] : d2;
                bi[j] = shift ? bi[j - 1] : idx;
            }
        }
        if (bd[0] > d2) { bd[0] = d2; bi[0] = idx; }
    }
}

// Tree-merge sorted 16-lists held in LDS down to thread 0's slot.
__device__ __forceinline__ void lds_tree_merge(float* md, int* mi, int t, int nthreads) {
    for (int s = 1; s < nthreads; s <<= 1) {
        __syncthreads();
        float rd[TOPK]; int ri[TOPK];
        bool m = ((t & (2 * s - 1)) == 0);
        if (m) {
            int a = t * TOPK, b = (t + s) * TOPK;
            int i = 0, j = 0;
            #pragma unroll
            for (int k = 0; k < TOPK; ++k) {   // i,j never exceed 15: safe
                float av = md[a + i], bv = md[b + j];
                bool ta = (av <= bv);
                rd[k] = ta ? av : bv;
                ri[k] = ta ? mi[a + i] : mi[b + j];
                if (ta) ++i; else ++j;
            }
        }
        __syncthreads();
        if (m) {
            #pragma unroll
            for (int k = 0; k < TOPK; ++k) { md[t * TOPK + k] = rd[k]; mi[t * TOPK + k] = ri[k]; }
        }
    }
    __syncthreads();
}

// Stage 1: stream all nodes with async double-buffered LDS tiles; squared L2
// distance via WMMA Gram-matrix diagonal; per-block top-16 to workspace.
__global__ void __launch_bounds__(K1_THREADS)
som_stage1(const float* __restrict__ q, const float* __restrict__ nodes,
           int nNodes, float* __restrict__ outD, int* __restrict__ outI) {
    __shared__ float qs[DIMS];
    __shared__ float slab[2][K1_WAVES][TILE * ROWPAD];
    __shared__ float md[K1_THREADS * TOPK];
    __shared__ int   mi[K1_THREADS * TOPK];

    const int t    = threadIdx.x;
    const int lane = t & 31;
    const int wave = t >> 5;

    if (t < DIMS) qs[t] = q[t];
    __syncthreads();

    float bd[TOPK]; int bi[TOPK];
    #pragma unroll
    for (int k = 0; k < TOPK; ++k) { bd[k] = FMAXV; bi[k] = 0; }

    const int nTiles     = (nNodes + TILE - 1) / TILE;
    const int waveGid    = blockIdx.x * K1_WAVES + wave;
    const int waveStride = gridDim.x * K1_WAVES;

    const int row  = lane & 15;           // node within tile this lane feeds to WMMA
    const int koff = (lane >> 4) << 1;    // K sub-offset per f32 16x16x4 A/B layout

    // Issue exactly 16 async b128 copies for a tile into slab[buf][wave].
    // OOB rows clamp to a valid address (results discarded later) so the
    // ASYNCcnt prefetch accounting stays exact.
    auto issue_tile = [&](int tile, int buf) {
        const int base = tile * TILE;
        wait_ds_0();   // WAR: DS reads of this slab must drain before async writes
        #pragma unroll 4
        for (int i = 0; i < TILE; ++i) {
            int r = base + i;
            if (r >= nNodes) r = nNodes - 1;
            uint32_t dst = (uint32_t)(uintptr_t)&slab[buf][wave][i * ROWPAD + lane * 4];
            async_copy_b128(dst, nodes + (size_t)r * DIMS + lane * 4);
        }
    };

    int  tile  = waveGid;
    bool valid = (tile < nTiles);
    int  buf   = 0;
    if (valid) issue_tile(tile, buf);

    while (valid) {
        const int  next   = tile + waveStride;
        const bool nvalid = (next < nTiles);
        if (nvalid) { issue_tile(next, buf ^ 1); wait_async_le16(); }
        else        { wait_async_0(); }

        // Gram of diff: acc[m][n] += sum_k (x_m[k]-q[k])*(x_n[k]-q[k]).
        // Same VGPR pair is both SRC0 (A 16x4) and SRC1 (B 4x16): identical lane layout.
        v8f acc = {};
        const float* rowp = &slab[buf][wave][row * ROWPAD + koff];
        const float* qp   = &qs[koff];
        #pragma unroll 8
        for (int kb = 0; kb < DIMS; kb += 4) {
            v2f x  = *(const v2f*)(rowp + kb);
            v2f qq = *(const v2f*)(qp + kb);
            v2f d  = x - qq;
            acc = __builtin_amdgcn_wmma_f32_16x16x4_f32(
                      false, d, false, d, (short)0, acc, false, false);
        }

        // Diagonal = squared distance. C/D layout: lanes 0-7 -> n=lane at c[lane];
        // lanes 24-31 -> n=lane-16 at c[lane-24]. Other lanes carry no diagonal.
        const int base = tile * TILE;
        float d2 = FMAXV; int gi = 0;
        if (lane < 8 || lane >= 24) {
            int sel = (lane < 8) ? lane : (lane - 24);
            #pragma unroll
            for (int j = 0; j < 8; ++j) if (sel == j) d2 = acc[j];
            gi = base + ((lane < 8) ? lane : (lane - 16));
            if (gi >= nNodes) d2 = FMAXV;
        }
        topk_insert(bd, bi, d2, gi);

        tile  = next;
        valid = nvalid;
        buf  ^= 1;
    }

    #pragma unroll
    for (int k = 0; k < TOPK; ++k) { md[t * TOPK + k] = bd[k]; mi[t * TOPK + k] = bi[k]; }
    lds_tree_merge(md, mi, t, K1_THREADS);

    if (t < TOPK) {
        outD[blockIdx.x * TOPK + t] = md[t];
        outI[blockIdx.x * TOPK + t] = mi[t];
    }
}

// Stage 2: single block reduces all per-block candidates to the global top-16.
__global__ void __launch_bounds__(K2_THREADS)
som_stage2(const float* __restrict__ cd, const int* __restrict__ ci,
           int nCand, float* __restrict__ out) {
    __shared__ float md[K2_THREADS * TOPK];
    __shared__ int   mi[K2_THREADS * TOPK];
    const int t = threadIdx.x;

    float bd[TOPK]; int bi[TOPK];
    #pragma unroll
    for (int k = 0; k < TOPK; ++k) { bd[k] = FMAXV; bi[k] = 0; }

    for (int c = t; c < nCand; c += K2_THREADS)
        topk_insert(bd, bi, cd[c], ci[c]);

    #pragma unroll
    for (int k = 0; k < TOPK; ++k) { md[t * TOPK + k] = bd[k]; mi[t * TOPK + k] = bi[k]; }
    lds_tree_merge(md, mi, t, K2_THREADS);

    if (t < TOPK) {
        out[t]        = (float)mi[t];   // indices (exact in f32: < 2^24)
        out[TOPK + t] = sqrtf(md[t]);   // distances (sqrt deferred to the final 16)
    }
}

extern "C" void kernel_launch(void* const* d_in, const int* in_sizes, int n_in,
                              void* d_out, int out_size, void* d_ws, size_t ws_size,
                              hipStream_t stream) {
    const float* q     = (const float*)d_in[0];
    const float* nodes = (const float*)d_in[1];
    const int nNodes   = in_sizes[1] / DIMS;

    float* wsD = (float*)d_ws;
    int*   wsI = (int*)((char*)d_ws + (size_t)K1_BLOCKS * TOPK * sizeof(float));

    som_stage1<<<K1_BLOCKS, K1_THREADS, 0, stream>>>(q, nodes, nNodes, wsD, wsI);
    som_stage2<<<1, K2_THREADS, 0, stream>>>(wsD, wsI, K1_BLOCKS * TOPK, (float*)d_out);
}